// AttentionBlock_88227218194824
// MI455X (gfx1250) — compile-verified
//
#include <hip/hip_runtime.h>
#include <hip/hip_bf16.h>
#include <math.h>

// ---------------------------------------------------------------------------
// AttentionBlock for MI455X (gfx1250, wave32, WMMA).
// Shapes: B=32, H=W=32 (N=1024 tokens), C=512, GROUPS=8, EPS=1e-3.
// All math in fp32 using V_WMMA_F32_16X16X4_F32 (keeps reference precision;
// the whole block is ~137 GFLOP vs ~130MB of HBM traffic -> memory bound,
// so the f32 matrix path is the right choice on this chip).
// ---------------------------------------------------------------------------

typedef __attribute__((ext_vector_type(2))) float v2f;
typedef __attribute__((ext_vector_type(8))) float v8f;

__device__ __forceinline__ v8f wmma_f32(v2f a, v2f b, v8f c) {
  // D(16x16,f32) += A(16x4,f32) * B(4x16,f32)
  return __builtin_amdgcn_wmma_f32_16x16x4_f32(
      /*neg_a=*/false, a, /*neg_b=*/false, b,
      /*c_mod=*/(short)0, c, /*reuse_a=*/false, /*reuse_b=*/false);
}

// ---------------------------------------------------------------------------
// Kernel 1: per-(batch,group) mean / rstd.  grid=256 blocks of 256 threads.
// Each (b,g) reduces 32*32*64 = 65536 elements.
// ---------------------------------------------------------------------------
__global__ __launch_bounds__(256) void gn_stats_kernel(
    const float* __restrict__ x, float* __restrict__ mean,
    float* __restrict__ rstd) {
  __shared__ float ssum[256];
  __shared__ float ssq[256];
  const int bg = blockIdx.x;            // 0..255
  const int b = bg >> 3, g = bg & 7;
  const int tid = threadIdx.x;
  const float* base = x + (size_t)b * 1024 * 512 + g * 64;
  float s = 0.f, q = 0.f;
  for (int i = tid; i < 65536; i += 256) {
    const int hw = i >> 6, cc = i & 63;
    const float v = base[(size_t)hw * 512 + cc];
    s += v;
    q += v * v;
  }
  ssum[tid] = s;
  ssq[tid] = q;
  __syncthreads();
  for (int off = 128; off > 0; off >>= 1) {
    if (tid < off) {
      ssum[tid] += ssum[tid + off];
      ssq[tid] += ssq[tid + off];
    }
    __syncthreads();
  }
  if (tid == 0) {
    const float m = ssum[0] * (1.0f / 65536.0f);
    const float var = ssq[0] * (1.0f / 65536.0f) - m * m;
    mean[bg] = m;
    rstd[bg] = rsqrtf(var + 1.0e-3f);
  }
}

// ---------------------------------------------------------------------------
// Kernel 2: h = (x - mean)*rstd*gamma + beta   (float4 vectorized elementwise)
// ---------------------------------------------------------------------------
__global__ __launch_bounds__(256) void gn_apply_kernel(
    const float* __restrict__ x, const float* __restrict__ gamma,
    const float* __restrict__ beta, const float* __restrict__ mean,
    const float* __restrict__ rstd, float* __restrict__ h) {
  const size_t idx = (size_t)blockIdx.x * blockDim.x + threadIdx.x;  // float4 id
  const size_t f = idx * 4;
  const int c = (int)(f & 511);
  const size_t row = f >> 9;            // token row (b*1024 + n)
  const int b = (int)(row >> 10);
  const int g = c >> 6;
  const float m = mean[b * 8 + g];
  const float r = rstd[b * 8 + g];
  const float4 xv = *(const float4*)(x + f);
  const float4 gv = *(const float4*)(gamma + c);
  const float4 bv = *(const float4*)(beta + c);
  float4 o;
  o.x = (xv.x - m) * r * gv.x + bv.x;
  o.y = (xv.y - m) * r * gv.y + bv.y;
  o.z = (xv.z - m) * r * gv.z + bv.z;
  o.w = (xv.w - m) * r * gv.w + bv.w;
  *(float4*)(h + f) = o;
}

// ---------------------------------------------------------------------------
// Kernel 3: QKV GEMM.  out(32768x1536) = h(32768x512) @ w(512x1536) + bias.
// Block = 8 waves, wave-grid 2x4, each wave owns a 32x64 tile (2x4 WMMA frags).
// grid = dim3(1536/256, 32768/64) = (6, 512).
// ---------------------------------------------------------------------------
__global__ __launch_bounds__(256) void qkv_gemm_kernel(
    const float* __restrict__ h, const float* __restrict__ w,
    const float* __restrict__ bias, float* __restrict__ out) {
  const int lane = threadIdx.x & 31;
  const int wv = threadIdx.x >> 5;             // 0..7
  const int l16 = lane & 15;
  const int klo = (lane >> 4) << 1;            // 0 | 2 (A/B K-pair per half-wave)
  const int hi8 = (lane >> 4) << 3;            // 0 | 8 (C/D row offset)
  const int m0 = blockIdx.y * 64 + (wv >> 2) * 32;
  const int n0 = blockIdx.x * 256 + (wv & 3) * 64;

  v8f acc[2][4];
  for (int i = 0; i < 2; ++i)
    for (int t = 0; t < 4; ++t) acc[i][t] = (v8f){};

  const float* arow0 = h + (size_t)(m0 + l16) * 512;
  const float* arow1 = arow0 + (size_t)16 * 512;
  for (int k = 0; k < 512; k += 4) {
    const v2f a0 = *(const v2f*)(arow0 + k + klo);
    const v2f a1 = *(const v2f*)(arow1 + k + klo);
    const float* brow = w + (size_t)(k + klo) * 1536 + n0 + l16;
    v2f b0, b1, b2, b3;
    b0.x = brow[0];  b0.y = brow[1536];
    b1.x = brow[16]; b1.y = brow[16 + 1536];
    b2.x = brow[32]; b2.y = brow[32 + 1536];
    b3.x = brow[48]; b3.y = brow[48 + 1536];
    acc[0][0] = wmma_f32(a0, b0, acc[0][0]);
    acc[0][1] = wmma_f32(a0, b1, acc[0][1]);
    acc[0][2] = wmma_f32(a0, b2, acc[0][2]);
    acc[0][3] = wmma_f32(a0, b3, acc[0][3]);
    acc[1][0] = wmma_f32(a1, b0, acc[1][0]);
    acc[1][1] = wmma_f32(a1, b1, acc[1][1]);
    acc[1][2] = wmma_f32(a1, b2, acc[1][2]);
    acc[1][3] = wmma_f32(a1, b3, acc[1][3]);
  }
  for (int t = 0; t < 4; ++t) {
    const int col = n0 + t * 16 + l16;
    const float bv = bias[col];
    for (int r = 0; r < 8; ++r) {
      out[(size_t)(m0 + r + hi8) * 1536 + col] = acc[0][t][r] + bv;
      out[(size_t)(m0 + 16 + r + hi8) * 1536 + col] = acc[1][t][r] + bv;
    }
  }
}

// ---------------------------------------------------------------------------
// Kernel 4: fused attention per (batch, 32-query tile).
//   S = (Q K^T), softmax(scale*S) in LDS, O = P V.   grid = dim3(32, 32).
// LDS score buffer 32 x 1028 floats (padded stride so half-wave rows m and
// m+8 land on disjoint banks).  qkv layout: row (b*1024+n), q=[0,512),
// k=[512,1024), v=[1024,1536).
// ---------------------------------------------------------------------------
#define SROW 1028
__global__ __launch_bounds__(256) void attn_kernel(
    const float* __restrict__ qkv, float* __restrict__ out) {
  __shared__ float s[32 * SROW];                 // 131584 bytes
  const int b = blockIdx.y;
  const int m0 = blockIdx.x * 32;
  const int lane = threadIdx.x & 31;
  const int wv = threadIdx.x >> 5;
  const int l16 = lane & 15;
  const int klo = (lane >> 4) << 1;
  const int hi8 = (lane >> 4) << 3;
  const float* base = qkv + (size_t)b * 1024 * 1536;

  // ---- Phase 1: raw scores S(32x1024) = Q(32x512) * K^T --------------------
  // 128 16x16 tiles, 16 per wave; both A and B are contiguous v2f loads.
  for (int tt = wv; tt < 128; tt += 8) {
    const int ti = tt >> 6;        // 0..1  (query sub-tile)
    const int tj = tt & 63;        // 0..63 (key tile)
    const float* qrow = base + (size_t)(m0 + ti * 16 + l16) * 1536;
    const float* krow = base + (size_t)(tj * 16 + l16) * 1536 + 512;
    v8f acc = (v8f){};
    for (int c = 0; c < 512; c += 4) {
      const v2f a = *(const v2f*)(qrow + c + klo);
      const v2f bb = *(const v2f*)(krow + c + klo);
      acc = wmma_f32(a, bb, acc);
    }
    float* srow = s + (size_t)(ti * 16 + hi8) * SROW + tj * 16 + l16;
    for (int r = 0; r < 8; ++r) srow[(size_t)r * SROW] = acc[r];
  }
  __syncthreads();

  // ---- Phase 2: softmax(scale * S) row-wise, 8 lanes per row ---------------
  {
    const float scale = 0.04419417382415922f;    // 1/sqrt(512)
    const int row = wv * 4 + (lane >> 3);        // 0..31
    const int t = lane & 7;
    float* sr = s + (size_t)row * SROW;
    float mx = -3.4e38f;
    for (int j = t; j < 1024; j += 8) mx = fmaxf(mx, sr[j]);
    mx = fmaxf(mx, __shfl_xor(mx, 1));
    mx = fmaxf(mx, __shfl_xor(mx, 2));
    mx = fmaxf(mx, __shfl_xor(mx, 4));
    float sum = 0.f;
    for (int j = t; j < 1024; j += 8) {
      const float e = __expf((sr[j] - mx) * scale);
      sr[j] = e;
      sum += e;
    }
    sum += __shfl_xor(sum, 1);
    sum += __shfl_xor(sum, 2);
    sum += __shfl_xor(sum, 4);
    const float inv = 1.0f / sum;
    for (int j = t; j < 1024; j += 8) sr[j] *= inv;
  }
  __syncthreads();

  // ---- Phase 3: O(32x512) = P(32x1024, LDS) * V(1024x512) ------------------
  // 64 16x16 tiles, 8 per wave; A from LDS (v2f), B strided dword pairs.
  for (int tt = wv; tt < 64; tt += 8) {
    const int ti = tt >> 5;        // 0..1
    const int tj = tt & 31;        // 0..31 (channel tile)
    const float* prow = s + (size_t)(ti * 16 + l16) * SROW;
    const float* vcol = base + 1024 + tj * 16 + l16;  // V(key,c)=base[key*1536+1024+c]
    v8f acc = (v8f){};
    for (int kk = 0; kk < 1024; kk += 4) {
      const v2f a = *(const v2f*)(prow + kk + klo);
      v2f bb;
      bb.x = vcol[(size_t)(kk + klo) * 1536];
      bb.y = vcol[(size_t)(kk + klo + 1) * 1536];
      acc = wmma_f32(a, bb, acc);
    }
    float* orow =
        out + (size_t)(b * 1024 + m0 + ti * 16 + hi8) * 512 + tj * 16 + l16;
    for (int r = 0; r < 8; ++r) orow[(size_t)r * 512] = acc[r];
  }
}

// ---------------------------------------------------------------------------
// Kernel 5: projection GEMM + bias + residual.
//   out(32768x512) = x + attn(32768x512) @ w_proj(512x512) + b_proj
// grid = dim3(512/256, 32768/64) = (2, 512).
// ---------------------------------------------------------------------------
__global__ __launch_bounds__(256) void proj_gemm_kernel(
    const float* __restrict__ h, const float* __restrict__ w,
    const float* __restrict__ bias, const float* __restrict__ x,
    float* __restrict__ out) {
  const int lane = threadIdx.x & 31;
  const int wv = threadIdx.x >> 5;
  const int l16 = lane & 15;
  const int klo = (lane >> 4) << 1;
  const int hi8 = (lane >> 4) << 3;
  const int m0 = blockIdx.y * 64 + (wv >> 2) * 32;
  const int n0 = blockIdx.x * 256 + (wv & 3) * 64;

  v8f acc[2][4];
  for (int i = 0; i < 2; ++i)
    for (int t = 0; t < 4; ++t) acc[i][t] = (v8f){};

  const float* arow0 = h + (size_t)(m0 + l16) * 512;
  const float* arow1 = arow0 + (size_t)16 * 512;
  for (int k = 0; k < 512; k += 4) {
    const v2f a0 = *(const v2f*)(arow0 + k + klo);
    const v2f a1 = *(const v2f*)(arow1 + k + klo);
    const float* brow = w + (size_t)(k + klo) * 512 + n0 + l16;
    v2f b0, b1, b2, b3;
    b0.x = brow[0];  b0.y = brow[512];
    b1.x = brow[16]; b1.y = brow[16 + 512];
    b2.x = brow[32]; b2.y = brow[32 + 512];
    b3.x = brow[48]; b3.y = brow[48 + 512];
    acc[0][0] = wmma_f32(a0, b0, acc[0][0]);
    acc[0][1] = wmma_f32(a0, b1, acc[0][1]);
    acc[0][2] = wmma_f32(a0, b2, acc[0][2]);
    acc[0][3] = wmma_f32(a0, b3, acc[0][3]);
    acc[1][0] = wmma_f32(a1, b0, acc[1][0]);
    acc[1][1] = wmma_f32(a1, b1, acc[1][1]);
    acc[1][2] = wmma_f32(a1, b2, acc[1][2]);
    acc[1][3] = wmma_f32(a1, b3, acc[1][3]);
  }
  for (int t = 0; t < 4; ++t) {
    const int col = n0 + t * 16 + l16;
    const float bv = bias[col];
    for (int r = 0; r < 8; ++r) {
      const size_t i0 = (size_t)(m0 + r + hi8) * 512 + col;
      const size_t i1 = (size_t)(m0 + 16 + r + hi8) * 512 + col;
      out[i0] = x[i0] + acc[0][t][r] + bv;
      out[i1] = x[i1] + acc[1][t][r] + bv;
    }
  }
}

// ---------------------------------------------------------------------------
// kernel_launch: orchestrate the 5 stages on `stream`.
// Workspace layout (floats): [mean 256][rstd 256][h 16.78M][qkv 50.33M]
//   -> ~268.4 MB.  attn output aliases the h buffer (h is dead after QKV).
// ---------------------------------------------------------------------------
extern "C" void kernel_launch(void* const* d_in, const int* in_sizes, int n_in,
                              void* d_out, int out_size, void* d_ws,
                              size_t ws_size, hipStream_t stream) {
  const float* x      = (const float*)d_in[0];
  const float* gamma  = (const float*)d_in[1];
  const float* beta   = (const float*)d_in[2];
  const float* w_qkv  = (const float*)d_in[3];
  const float* b_qkv  = (const float*)d_in[4];
  const float* w_proj = (const float*)d_in[5];
  const float* b_proj = (const float*)d_in[6];
  float* out = (float*)d_out;

  float* wsf  = (float*)d_ws;
  float* mean = wsf;                       // 256
  float* rstd = wsf + 256;                 // 256
  float* h    = wsf + 512;                 // 32768*512  = 16777216
  float* qkv  = h + (size_t)16777216;      // 32768*1536 = 50331648
  float* attn = h;                         // alias: h dead after QKV GEMM

  gn_stats_kernel<<<256, 256, 0, stream>>>(x, mean, rstd);
  gn_apply_kernel<<<16384, 256, 0, stream>>>(x, gamma, beta, mean, rstd, h);
  qkv_gemm_kernel<<<dim3(6, 512), 256, 0, stream>>>(h, w_qkv, b_qkv, qkv);
  attn_kernel<<<dim3(32, 32), 256, 0, stream>>>(qkv, attn);
  proj_gemm_kernel<<<dim3(2, 512), 256, 0, stream>>>(attn, w_proj, b_proj, x,
                                                     out);
}